// RGCN_73538430042254
// MI455X (gfx1250) — compile-verified
//
#include <hip/hip_runtime.h>

#define DIM 64

typedef float v2f __attribute__((ext_vector_type(2)));
typedef float v8f __attribute__((ext_vector_type(8)));

// out = [RELU?](X)[job*32 .. +32) @ Wm (64x64 row-major) [+ BIAS?]
// One wave handles TWO 16-row tiles x full 64 cols: B fragments are reused
// across both row tiles (8 WMMA per 10 VMEM per k-step), 128 WMMAs per wave.
// RELU/BIAS are compile-time so the unrolled k-loop has zero control flow.
// fp32 WMMA: V_WMMA_F32_16X16X4_F32 (wave32).
//   A frag (16x4): lane%16 = M row; lanes 0-15: v0=K0,v1=K1; lanes 16-31: v0=K2,v1=K3.
//   B frag (4x16): lane%16 = N col; lanes 0-15: v0=K0,v1=K1; lanes 16-31: v0=K2,v1=K3.
//   C/D (16x16):   VGPR r -> row = r + 8*(lane/16), col = lane%16.
template <bool RELU, bool BIAS>
__global__ __launch_bounds__(128) void rgcn_gemm64(
    const float* __restrict__ X, const float* __restrict__ Wm,
    const float* __restrict__ bias, float* __restrict__ out, int nrows)
{
    const int lane = threadIdx.x & 31;
    const int wave = threadIdx.x >> 5;
    const int job  = blockIdx.x * 4 + wave;            // each job = 32 rows
    const int row0 = job * 32;
    if (row0 >= nrows) return;                         // wave-uniform exit (nrows % 32 == 0)
    const int half = lane >> 4;                        // 0: K pair {0,1}; 1: K pair {2,3}
    const int l16  = lane & 15;

    v8f acc[2][4];
    #pragma unroll
    for (int s = 0; s < 2; ++s)
        #pragma unroll
        for (int t = 0; t < 4; ++t)
            acc[s][t] = v8f{};

    const float* xrow0 = X + (size_t)(row0 + l16) * DIM + 2 * half;
    const float* xrow1 = xrow0 + 16 * DIM;

    #pragma unroll
    for (int k0 = 0; k0 < DIM; k0 += 4) {
        v2f a0 = *(const v2f*)(xrow0 + k0);
        v2f a1 = *(const v2f*)(xrow1 + k0);
        if (RELU) {                                    // compile-time; fused activation
            a0.x = fmaxf(a0.x, 0.0f); a0.y = fmaxf(a0.y, 0.0f);
            a1.x = fmaxf(a1.x, 0.0f); a1.y = fmaxf(a1.y, 0.0f);
        }
        const float* wk = Wm + (size_t)(k0 + 2 * half) * DIM + l16;
        #pragma unroll
        for (int t = 0; t < 4; ++t) {
            v2f b;
            b.x = wk[t * 16];
            b.y = wk[t * 16 + DIM];
            acc[0][t] = __builtin_amdgcn_wmma_f32_16x16x4_f32(
                false, a0, false, b, (short)0, acc[0][t], false, false);
            acc[1][t] = __builtin_amdgcn_wmma_f32_16x16x4_f32(
                false, a1, false, b, (short)0, acc[1][t], false, false);
        }
    }

    #pragma unroll
    for (int s = 0; s < 2; ++s) {
        const int rbase = row0 + s * 16 + 8 * half;
        #pragma unroll
        for (int t = 0; t < 4; ++t) {
            const int col = t * 16 + l16;
            const float bv = BIAS ? bias[col] : 0.0f;
            #pragma unroll
            for (int r = 0; r < 8; ++r) {
                out[(size_t)(rbase + r) * DIM + col] = acc[s][t][r] + bv;
            }
        }
    }
}

// One edge handled by 16 threads; each thread moves 4 consecutive floats.
// msg = (etype==0 ? H0 : H1)[src];  out[dst] += msg  (atomic, fp32).
__global__ __launch_bounds__(256) void rgcn_scatter(
    const float* __restrict__ H0, const float* __restrict__ H1,
    const int* __restrict__ src, const int* __restrict__ dst,
    const int* __restrict__ etype, float* __restrict__ out, int nEdges)
{
    const int tid = blockIdx.x * blockDim.x + threadIdx.x;
    const int e = tid >> 4;
    const int q = tid & 15;
    if (e >= nEdges) return;
    const float* h = (etype[e] == 0 ? H0 : H1) + (size_t)src[e] * DIM + q * 4;
    const float4 m = *(const float4*)h;
    float* o = out + (size_t)dst[e] * DIM + q * 4;
    atomicAdd(o + 0, m.x);
    atomicAdd(o + 1, m.y);
    atomicAdd(o + 2, m.z);
    atomicAdd(o + 3, m.w);
}

extern "C" void kernel_launch(void* const* d_in, const int* in_sizes, int n_in,
                              void* d_out, int out_size, void* d_ws, size_t ws_size,
                              hipStream_t stream)
{
    (void)n_in; (void)out_size; (void)ws_size;

    // setup_inputs order:
    // 0 features [N,64] f32 | 1 src [E] i32 | 2 dst [E] i32 | 3 etype [E] i32
    // 4 W0 [2,64,64] | 5 Ws0 [64,64] | 6 b0 [64]
    // 7 W1            | 8 Ws1         | 9 b1
    // 10 W2           | 11 Ws2        | 12 b2
    const float* features = (const float*)d_in[0];
    const int*   src      = (const int*)d_in[1];
    const int*   dst      = (const int*)d_in[2];
    const int*   ety      = (const int*)d_in[3];
    const float* W[3]  = {(const float*)d_in[4], (const float*)d_in[7],  (const float*)d_in[10]};
    const float* Ws[3] = {(const float*)d_in[5], (const float*)d_in[8],  (const float*)d_in[11]};
    const float* b[3]  = {(const float*)d_in[6], (const float*)d_in[9],  (const float*)d_in[12]};

    const int N = in_sizes[0] / DIM;    // 100000 (divisible by 32)
    const int E = in_sizes[1];          // 1000000

    const size_t NS = (size_t)N * DIM;  // floats per node buffer (25.6 MB)
    float* H0   = (float*)d_ws;
    float* H1   = H0 + NS;
    float* bufA = H0 + 2 * NS;
    float* bufB = H0 + 3 * NS;

    const int jobs = (N + 31) / 32;     // 32-row jobs, one wave each
    dim3 gGrid((jobs + 3) / 4);         // 4 waves per block
    dim3 gBlk(128);
    dim3 sGrid((unsigned)(((size_t)E * 16 + 255) / 256));

    const float* x = features;
    float* outs[3] = {bufA, bufB, (float*)d_out};

    for (int L = 0; L < 3; ++L) {
        float* o = outs[L];
        if (L == 0) {
            // First layer: raw features, no fused activation
            rgcn_gemm64<false, false><<<gGrid, gBlk, 0, stream>>>(x, W[L],             nullptr, H0, N);
            rgcn_gemm64<false, false><<<gGrid, gBlk, 0, stream>>>(x, W[L] + DIM * DIM, nullptr, H1, N);
            rgcn_gemm64<false, true ><<<gGrid, gBlk, 0, stream>>>(x, Ws[L],            b[L],    o,  N);
        } else {
            // Later layers: previous layer's ReLU fused into A-fragment loads
            rgcn_gemm64<true, false><<<gGrid, gBlk, 0, stream>>>(x, W[L],             nullptr, H0, N);
            rgcn_gemm64<true, false><<<gGrid, gBlk, 0, stream>>>(x, W[L] + DIM * DIM, nullptr, H1, N);
            rgcn_gemm64<true, true ><<<gGrid, gBlk, 0, stream>>>(x, Ws[L],            b[L],    o,  N);
        }
        // Edge-wise message scatter-add (pre-activation stored; next layer applies ReLU)
        rgcn_scatter<<<sGrid, 256, 0, stream>>>(H0, H1, src, dst, ety, o, E);
        x = o;
    }
}